// RelationalGNN_43078521978922
// MI455X (gfx1250) — compile-verified
//
#include <hip/hip_runtime.h>

// RGCN (4 layers) for gfx1250: fp32 WMMA 16x16x4, async-LDS double-buffered A tiles.
// Layer: agg[r,i,:] = mean_{j in N_r(i)} h[j,:]; out = sum_r agg[r]@W[r] + h@root + b; ReLU.

#define NN  50000       // nodes
#define NE  800000      // edges
#define NR  8           // relations
#define DIN 128
#define DH0 256
#define DH1 64

typedef float v2f __attribute__((ext_vector_type(2)));
typedef float v8f __attribute__((ext_vector_type(8)));

// ---------------- utility kernels ----------------

__global__ void zero_kernel(float* __restrict__ p, long n) {
  long i = (long)blockIdx.x * blockDim.x + threadIdx.x;
  long stride = (long)gridDim.x * blockDim.x;
  for (; i < n; i += stride) p[i] = 0.0f;
}

// per-(relation,dst) edge counts (static across layers)
__global__ void count_kernel(const int* __restrict__ etype,
                             const int* __restrict__ dst,
                             float* __restrict__ cnt) {
  int e = blockIdx.x * blockDim.x + threadIdx.x;
  if (e < NE) {
    int seg = etype[e] * NN + dst[e];
    atomicAdd(&cnt[seg], 1.0f);
  }
}

// scatter-add source features into agg[(r*N+dst), :]
__global__ void scatter_kernel(const float* __restrict__ x,
                               const int* __restrict__ src,
                               const int* __restrict__ dst,
                               const int* __restrict__ etype,
                               float* __restrict__ agg,
                               int kin, int kshift) {
  long total = (long)NE << kshift;
  long i = (long)blockIdx.x * blockDim.x + threadIdx.x;
  long stride = (long)gridDim.x * blockDim.x;
  int kmask = kin - 1;
  for (; i < total; i += stride) {
    int e = (int)(i >> kshift);
    int f = (int)(i & kmask);
    int seg = etype[e] * NN + dst[e];
    atomicAdd(&agg[((long)seg << kshift) + f], x[((long)src[e] << kshift) + f]);
  }
}

// agg[seg, f] /= max(cnt[seg], 1)
__global__ void normalize_kernel(float* __restrict__ agg,
                                 const float* __restrict__ cnt,
                                 int kshift) {
  long total = ((long)NR * NN) << kshift;
  long i = (long)blockIdx.x * blockDim.x + threadIdx.x;
  long stride = (long)gridDim.x * blockDim.x;
  for (; i < total; i += stride) {
    long seg = i >> kshift;
    float c = cnt[seg];
    agg[i] *= __frcp_rn(fmaxf(c, 1.0f));
  }
}

// ---------------- fused RGCN GEMM (WMMA f32 16x16x4) ----------------
// block = 128 threads (4 waves). Block computes a 16-row x 64-col output tile;
// each wave owns one 16x16 column tile. The 9 A tiles (8 relations + root,
// each 16 x kin) are copied global->LDS with GLOBAL_LOAD_ASYNC_TO_LDS_B128 and
// double-buffered so tile r+1 streams in while tile r feeds the WMMA k-loop.
// B (W[r] / root, <=2MB per layer) streams from global and stays L2-resident.

#define LDS_PAD 4
#define MAX_LDA (DH0 + LDS_PAD)

__global__ __launch_bounds__(128)
void rgcn_gemm_kernel(const float* __restrict__ agg,   // [R, N, kin] (normalized)
                      const float* __restrict__ xin,   // [N, kin]
                      const float* __restrict__ W,     // [R, kin, kout]
                      const float* __restrict__ root,  // [kin, kout]
                      const float* __restrict__ bias,  // [kout]
                      float* __restrict__ out,         // [N, kout]
                      int kin, int kout, int do_relu) {
  __shared__ float lds_a[2][16 * MAX_LDA];
  const int ldA = kin + LDS_PAD;   // row stride (floats); 16B alignment preserved

  const int tid  = threadIdx.x;
  const int wave = tid >> 5;
  const int lane = tid & 31;
  const int mn   = lane & 15;        // M for A-frag, N for B-frag, col for C
  const int kh   = (lane >> 4) * 2;  // K sub-offset 0 or 2 (A/B frag layout)

  const int rowBase = blockIdx.x * 16;
  const int colBase = blockIdx.y * 64 + wave * 16;
  const int col     = colBase + mn;

  const int vecsPerRow = kin >> 2;            // float4 per A row
  const int totalVecs  = 16 * vecsPerRow;     // multiple of 128 -> uniform EXEC

  // A-tile source pointer for "relation" r (r == NR is the root term on xin)
  auto a_ptr = [&](int r) -> const float* {
    return (r < NR) ? (agg + ((long)r * NN + rowBase) * kin)
                    : (xin + (long)rowBase * kin);
  };

  // async global -> LDS copy of one 16 x kin tile (B128 per lane per issue)
  auto issue_tile = [&](const float* Asrc, int buf) {
    for (int idx = tid; idx < totalVecs; idx += 128) {
      int row = idx / vecsPerRow;
      int c4  = idx - row * vecsPerRow;
      unsigned ldsoff = (unsigned)(unsigned long long)(const void*)
                          &lds_a[buf][row * ldA + c4 * 4];
      unsigned long long gaddr = (unsigned long long)(const void*)
                          (Asrc + (long)row * kin + c4 * 4);
      asm volatile("global_load_async_to_lds_b128 %0, %1, off"
                   :: "v"(ldsoff), "v"(gaddr) : "memory");
    }
  };

  v8f acc = {};

  issue_tile(a_ptr(0), 0);  // prologue: stage tile 0

  for (int r = 0; r <= NR; ++r) {
    asm volatile("s_wait_asynccnt 0" ::: "memory");  // this wave's copies landed
    __syncthreads();                                 // all waves' copies visible
    if (r < NR) issue_tile(a_ptr(r + 1), (r + 1) & 1);  // overlap next tile

    const float* Bsrc = (r < NR) ? (W + (long)r * kin * kout) : root;
    const float* abuf = lds_a[r & 1];

    for (int k0 = 0; k0 < kin; k0 += 4) {
      v2f a, b;
      // A fragment: 16x4 f32, lane m = mn, K = k0 + kh + {0,1}
      a.x = abuf[mn * ldA + k0 + kh + 0];
      a.y = abuf[mn * ldA + k0 + kh + 1];
      // B fragment: 4x16 f32, lane n = mn, K = k0 + kh + {0,1}
      const float* bp = Bsrc + (long)(k0 + kh) * kout + col;
      b.x = bp[0];
      b.y = bp[kout];
      acc = __builtin_amdgcn_wmma_f32_16x16x4_f32(
          /*neg_a=*/false, a, /*neg_b=*/false, b,
          /*c_mod=*/(short)0, acc, /*reuse_a=*/false, /*reuse_b=*/false);
    }
  }

  // C/D layout: VGPR v -> M = v (lanes 0-15) or v+8 (lanes 16-31), N = lane&15
  const float bval = bias[col];
  const int rbase  = rowBase + ((lane >> 4) << 3);
#pragma unroll
  for (int v = 0; v < 8; ++v) {
    float val = acc[v] + bval;
    if (do_relu) val = fmaxf(val, 0.0f);
    out[(long)(rbase + v) * kout + col] = val;
  }
}

// ---------------- host orchestration ----------------

extern "C" void kernel_launch(void* const* d_in, const int* in_sizes, int n_in,
                              void* d_out, int out_size, void* d_ws, size_t ws_size,
                              hipStream_t stream) {
  (void)in_sizes; (void)n_in; (void)out_size; (void)ws_size;

  const float* x     = (const float*)d_in[0];
  const int*   eidx  = (const int*)d_in[1];   // [2, E]
  const int*   etype = (const int*)d_in[2];
  const float* W1 = (const float*)d_in[3],  *r1 = (const float*)d_in[4],  *b1 = (const float*)d_in[5];
  const float* W2 = (const float*)d_in[6],  *r2 = (const float*)d_in[7],  *b2 = (const float*)d_in[8];
  const float* W3 = (const float*)d_in[9],  *r3 = (const float*)d_in[10], *b3 = (const float*)d_in[11];
  const float* W4 = (const float*)d_in[12], *r4 = (const float*)d_in[13], *b4 = (const float*)d_in[14];
  float* out = (float*)d_out;

  const int* src = eidx;       // row 0
  const int* dst = eidx + NE;  // row 1

  // workspace layout
  float* agg = (float*)d_ws;                               // R*N*256
  float* cnt = agg + (size_t)NR * NN * DH0;                // R*N
  float* hA  = cnt + (size_t)NR * NN;                      // N*256
  float* hB  = hA + (size_t)NN * DH0;                      // N*256

  // degree counts: static for all layers
  zero_kernel<<<2048, 256, 0, stream>>>(cnt, (long)NR * NN);
  count_kernel<<<(NE + 255) / 256, 256, 0, stream>>>(etype, dst, cnt);

  auto layer = [&](const float* hin, int kin, int kshift,
                   const float* W, const float* rt, const float* b,
                   float* hout, int kout, int relu) {
    zero_kernel<<<4096, 256, 0, stream>>>(agg, ((long)NR * NN) << kshift);
    scatter_kernel<<<8192, 256, 0, stream>>>(hin, src, dst, etype, agg, kin, kshift);
    normalize_kernel<<<4096, 256, 0, stream>>>(agg, cnt, kshift);
    dim3 grid(NN / 16, kout / 64);
    rgcn_gemm_kernel<<<grid, 128, 0, stream>>>(agg, hin, W, rt, b, hout, kin, kout, relu);
  };

  layer(x,  DIN, 7, W1, r1, b1, hA,  DH0, 1);
  layer(hA, DH0, 8, W2, r2, b2, hB,  DH0, 1);
  layer(hB, DH0, 8, W3, r3, b3, hA,  DH0, 1);
  layer(hA, DH0, 8, W4, r4, b4, out, DH1, 0);
}